// EmbeddingsLayer_57028575756670
// MI455X (gfx1250) — compile-verified
//
#include <hip/hip_runtime.h>

// Problem constants from the reference.
#define B_      256
#define S_      2048
#define NV_     64
#define ST_     17
#define BL_     5
#define BOARD_F (ST_ * BL_ * 3)      // 255 floats of boards per batch
#define OUTW    (BOARD_F + S_)       // 2303 floats per batch row

// gfx1250 async global->LDS path, guarded so host pass / older toolchains
// still compile (falls back to plain coalesced loads).
#if defined(__gfx1250__) && __has_builtin(__builtin_amdgcn_global_load_async_to_lds_b128)
#define HAVE_ASYNC_LDS 1
#else
#define HAVE_ASYNC_LDS 0
#endif

#if HAVE_ASYNC_LDS
typedef int v4i_t __attribute__((ext_vector_type(4)));
typedef __attribute__((address_space(1))) v4i_t* gptr_v4i;
typedef __attribute__((address_space(3))) v4i_t* lptr_v4i;
#endif

// ---------------------------------------------------------------------------
// Kernel 1: boards. One 256-thread block (8 wave32 waves) per batch.
// Stable truncated bucket-compaction: per-thread chunk counts -> per-style
// exclusive scan across threads -> replay chunk, write rows with rank < 5.
// ---------------------------------------------------------------------------
__global__ __launch_bounds__(256) void boards_kernel(
    const float* __restrict__ seq,   // [B,S,3]
    const int*   __restrict__ mi,    // [B,S]
    const int*   __restrict__ cnt,   // [B]
    float*       __restrict__ out)   // [B,OUTW]
{
    // 257 padding: prefix-scan phase has 17 threads striding rows; padding
    // spreads them across the 64 LDS banks.
    __shared__ int s_cnt[ST_][257];
    __shared__ int s_tl;

    const int b = blockIdx.x;
    const int t = threadIdx.x;

    if (t == 0) s_tl = mi[(size_t)b * S_ + (cnt[b] - 1)] + 1;

    #pragma unroll
    for (int c = 0; c < ST_; ++c) s_cnt[c][t] = 0;
    __syncthreads();

    const int tl   = s_tl;
    const int base = t * 8;                         // 256 threads * 8 = 2048 = S
    const float* __restrict__ srow0 = seq + ((size_t)b * S_ + base) * 3;
    __builtin_prefetch(srow0, 0, 0);                // global_prefetch_b8

    int mloc[8];
    #pragma unroll
    for (int i = 0; i < 8; ++i) {
        const int j = base + i;
        int mv = -1;
        if (j < tl) mv = (int)srow0[i * 3 + 2];     // mode column, exact small int
        mloc[i] = mv;
        if (mv >= 0) s_cnt[mv][t] += 1;             // thread-private LDS cell
    }
    __syncthreads();

    // Per-style exclusive prefix over the 256 per-thread counts.
    if (t < ST_) {
        int acc = 0;
        for (int i = 0; i < 256; ++i) {
            const int x = s_cnt[t][i];
            s_cnt[t][i] = acc;
            acc += x;
        }
    }
    // Zero-fill the board region (harness poisons d_out; unmatched slots = 0).
    float* __restrict__ ob = out + (size_t)b * OUTW;
    if (t < BOARD_F) ob[t] = 0.0f;
    __syncthreads();

    // Replay chunk: global rank = exclusive prefix + local occurrence order.
    #pragma unroll
    for (int i = 0; i < 8; ++i) {
        const int mv = mloc[i];
        if (mv >= 0) {
            const int r = s_cnt[mv][t];
            s_cnt[mv][t] = r + 1;
            if (r < BL_) {
                const int o = (mv * BL_ + r) * 3;
                ob[o + 0] = srow0[i * 3 + 0];
                ob[o + 1] = srow0[i * 3 + 1];
                ob[o + 2] = srow0[i * 3 + 2];
            }
        }
    }
}

// ---------------------------------------------------------------------------
// Kernel 2: pxvm. Grid (S/256, B). Stage the batch's sorted mi row (8 KB) in
// LDS (gfx1250 async DMA path, ASYNCcnt-tracked), then per-thread binary
// search in LDS and the v gather.
// ---------------------------------------------------------------------------
__global__ __launch_bounds__(256) void pxvm_kernel(
    const float* __restrict__ v,     // [B,S,NV]
    const int*   __restrict__ mi,    // [B,S] sorted per row
    const int*   __restrict__ n,     // [B,S]
    const int*   __restrict__ cnt,   // [B]
    float*       __restrict__ out)   // [B,OUTW]
{
    __shared__ int smi[S_];

    const int b = blockIdx.y;
    const int t = threadIdx.x;
    const int j = blockIdx.x * 256 + t;
    const int* __restrict__ mrow = mi + (size_t)b * S_;

#if HAVE_ASYNC_LDS
    {
        // Each lane DMAs 32 contiguous bytes: two async B128 transfers.
        gptr_v4i gsrc = (gptr_v4i)(mrow + t * 8);
        lptr_v4i ldst = (lptr_v4i)(smi + t * 8);
        __builtin_amdgcn_global_load_async_to_lds_b128(gsrc, ldst, 0, 0);
        __builtin_amdgcn_global_load_async_to_lds_b128(gsrc, ldst, 16, 0);
#if __has_builtin(__builtin_amdgcn_s_wait_asynccnt)
        __builtin_amdgcn_s_wait_asynccnt(0);
#else
        asm volatile("s_wait_asynccnt 0" ::: "memory");
#endif
    }
    __syncthreads();
#else
    #pragma unroll
    for (int i = 0; i < 8; ++i) smi[t + i * 256] = mrow[t + i * 256];
    __syncthreads();
#endif

    const int c  = cnt[b];           // uniform per block -> scalarized
    const int tl = smi[c - 1] + 1;

    float r = 0.0f;
    if (j < tl) {
        // lf = #{k in [1, c-1] : mi[b,k] <= j}  (searchsorted right, in LDS)
        int lo = 1, hi = c;
        while (lo < hi) {
            const int mid = (lo + hi) >> 1;
            if (smi[mid] <= j) lo = mid + 1; else hi = mid;
        }
        const int lf = lo - 1;
        const int nv = n[(size_t)b * S_ + j];
        r = v[((size_t)b * S_ + lf) * NV_ + nv];
    }
    out[(size_t)b * OUTW + BOARD_F + j] = r;
}

// ---------------------------------------------------------------------------
// Entry point. Input order (setup_inputs dict): seq, p, v, mi, n, cnt.
// p is never used by the reference output.
// ---------------------------------------------------------------------------
extern "C" void kernel_launch(void* const* d_in, const int* in_sizes, int n_in,
                              void* d_out, int out_size, void* d_ws, size_t ws_size,
                              hipStream_t stream) {
    const float* seq = (const float*)d_in[0];
    const float* v   = (const float*)d_in[2];
    const int*   mi  = (const int*)d_in[3];
    const int*   n   = (const int*)d_in[4];
    const int*   cnt = (const int*)d_in[5];
    float* out = (float*)d_out;

    boards_kernel<<<dim3(B_), dim3(256), 0, stream>>>(seq, mi, cnt, out);
    pxvm_kernel<<<dim3(S_ / 256, B_), dim3(256), 0, stream>>>(v, mi, n, cnt, out);

    (void)in_sizes; (void)n_in; (void)out_size; (void)d_ws; (void)ws_size;
}